// VQVAE_8873402433753
// MI455X (gfx1250) — compile-verified
//
#include <hip/hip_runtime.h>
#include <float.h>
#include <math.h>

// ---------------------------------------------------------------------------
// VQ-VAE codebook quantization for MI455X (gfx1250, wave32).
//
// The inner product fn[g,0:4] . embed[k,0:4] has K-dim exactly 4, the native
// K of V_WMMA_F32_16X16X4_F32: one WMMA = 16-group x 16-code dot tile,
// 64 WMMAs/wave cover all 1024 codes. Inner loop has NO cross-lane ops:
// each lane loads the full float4 code vector so ||e||^2 is lane-local.
// ---------------------------------------------------------------------------

typedef __attribute__((ext_vector_type(2))) float v2f;
typedef __attribute__((ext_vector_type(8))) float v8f;

#define D_TOT     512                 // feature dim per (b,p) row
#define VQD       4                   // vq_dim
#define GROUPS    (D_TOT / VQD)       // 128 groups per row
#define NCODES    1024                // K
#define ROWS      512                 // b*p = 32*16
#define OUT_ELEMS (ROWS * D_TOT)      // 262144

// FloatBiter: quantize scale to 8 bits in log4 space (forward value only).
__device__ __forceinline__ float float_biter(float x) {
    float xc = fminf(fmaxf(x + 1.0f, 1.0f), 16.0f);
    float lg = logf(xc) * 0.7213475204444817f;   // log(xc)/log(4), in [0,2]
    float s  = 0.0f;
    float db = 1.0f;                              // 2^i
#pragma unroll
    for (int i = 0; i < 8; ++i) {
        int bit = ((int)floorf(lg * db)) & 1;     // x>=0 so trunc==floor
        s += bit ? (1.0f / db) : 0.0f;
        db *= 2.0f;
    }
    return exp2f(2.0f * s) - 1.0f;                // 4^s - 1
}

// Kernel 1: per-row L2 norm + FloatBiter; one wave per row. Also zeroes the
// loss accumulators every call (harness does not re-poison between replays).
__global__ void vq_scale_kernel(const float* __restrict__ feedback,
                                float* __restrict__ scale,
                                float* __restrict__ scale_q,
                                float* __restrict__ loss_out) {
    const int lane = threadIdx.x & 31;
    const int wave = threadIdx.x >> 5;
    const int row  = blockIdx.x * 8 + wave;

    const float4* rp = reinterpret_cast<const float4*>(feedback + row * D_TOT);
    float acc = 0.0f;
#pragma unroll
    for (int j = 0; j < 4; ++j) {                 // 128 float4 per row
        float4 v = rp[lane + j * 32];
        acc += v.x * v.x + v.y * v.y + v.z * v.z + v.w * v.w;
    }
#pragma unroll
    for (int off = 1; off < 32; off <<= 1)
        acc += __shfl_xor(acc, off, 32);

    if (lane == 0) {
        float s = sqrtf(acc);
        scale[row]   = s;
        scale_q[row] = float_biter(s);
    }
    if (blockIdx.x == 0 && threadIdx.x == 0) {
        loss_out[0] = 0.0f;
        loss_out[1] = 0.0f;
    }
}

// Kernel 2: one block per (b,p) row; 8 waves, each wave owns 16 groups and
// scans all 1024 codes with 64 x V_WMMA_F32_16X16X4_F32.
__global__ void __launch_bounds__(256, 2)
vq_main_kernel(const float* __restrict__ feedback,
               const float* __restrict__ embed,
               const float* __restrict__ scale,
               const float* __restrict__ scale_q,
               float* __restrict__ out,
               float* __restrict__ loss_out) {
    const int row     = blockIdx.x;        // 0..511
    const int lane    = threadIdx.x & 31;
    const int wave    = threadIdx.x >> 5;  // 0..7
    const int halfSel = lane >> 4;         // 0: lanes 0-15, 1: lanes 16-31
    const int r       = lane & 15;

    const float inv_s = 1.0f / scale[row];
    const float sq    = scale_q[row];

    const float* rowp  = feedback + row * D_TOT;
    const int    gbase = wave * 16;        // first group of this wave

    // A fragment (16x4 f32): lanes 0-15 hold K={0,1} of row M=lane,
    // lanes 16-31 hold K={2,3} of row M=lane-16 (ISA 32-bit A layout).
    v2f a;
    {
        const int g = gbase + r;
        v2f av = *reinterpret_cast<const v2f*>(rowp + 4 * g + 2 * halfSel);
        a.x = av.x * inv_s;
        a.y = av.y * inv_s;
    }

    // Running argmin of score = ||e||^2 - 2*fn.e (same argmin as full dist).
    float mins[8];
    int   midx[8];
#pragma unroll
    for (int i = 0; i < 8; ++i) { mins[i] = FLT_MAX; midx[i] = 0; }

    const float4* embed4 = reinterpret_cast<const float4*>(embed);

#pragma unroll 4
    for (int t = 0; t < NCODES / 16; ++t) {
        const int n = t * 16 + r;          // this lane's code column N
        // Full code vector per lane: ||e||^2 is lane-local (no shuffle),
        // and the B fragment half is a register select, not a new load.
        float4 e = embed4[n];
        v2f b;
        b.x = halfSel ? e.z : e.x;
        b.y = halfSel ? e.w : e.y;
        float ee = e.x * e.x + e.y * e.y + e.z * e.z + e.w * e.w;

        v8f c = {};
        v8f d = __builtin_amdgcn_wmma_f32_16x16x4_f32(
            /*neg_a=*/false, a, /*neg_b=*/false, b,
            /*c_mod=*/(short)0, c, /*reuse_a=*/false, /*reuse_b=*/false);

        // D layout: lane holds N=r, VGPR i holds row M=i (+8 for upper half).
#pragma unroll
        for (int i = 0; i < 8; ++i) {
            float score = __builtin_fmaf(-2.0f, d[i], ee);
            if (score < mins[i]) { mins[i] = score; midx[i] = n; }
        }
    }

    // Cross-lane min per row (xor over the 16-lane half; index tie-break
    // mimics argmin's first-occurrence rule), then write output + loss.
    float lossLocal = 0.0f;
#pragma unroll
    for (int i = 0; i < 8; ++i) {
        float sc = mins[i];
        int   id = midx[i];
#pragma unroll
        for (int off = 1; off < 16; off <<= 1) {
            float osc = __shfl_xor(sc, off, 32);
            int   oid = __shfl_xor(id, off, 32);
            if (osc < sc || (osc == sc && oid < id)) { sc = osc; id = oid; }
        }
        if (r == i) {                       // 2 lanes/wave own row i (+8)
            const int m = halfSel * 8 + i;  // row M within the 16x16 tile
            const int g = gbase + m;        // group index in this row
            float4 e4 = embed4[id];
            float4 f4 = reinterpret_cast<const float4*>(rowp)[g];
            float dx = e4.x - f4.x * inv_s;
            float dy = e4.y - f4.y * inv_s;
            float dz = e4.z - f4.z * inv_s;
            float dw = e4.w - f4.w * inv_s;
            lossLocal += dx * dx + dy * dy + dz * dz + dw * dw;
            float4 o4 = make_float4(e4.x * sq, e4.y * sq, e4.z * sq, e4.w * sq);
            reinterpret_cast<float4*>(out + row * D_TOT)[g] = o4;
        }
    }

    // Wave-reduce the loss partials, one atomic per wave. Pre-divide so the
    // accumulator holds the mean; loss1 == loss2 in forward values.
#pragma unroll
    for (int off = 1; off < 32; off <<= 1)
        lossLocal += __shfl_xor(lossLocal, off, 32);
    if (lane == 0) {
        float contrib = lossLocal * (1.0f / (float)OUT_ELEMS);
        atomicAdd(loss_out + 0, contrib);
        atomicAdd(loss_out + 1, contrib);
    }
}

extern "C" void kernel_launch(void* const* d_in, const int* in_sizes, int n_in,
                              void* d_out, int out_size, void* d_ws, size_t ws_size,
                              hipStream_t stream) {
    const float* feedback = (const float*)d_in[0];  // [32,16,512]
    const float* embed    = (const float*)d_in[1];  // [1024,4]
    float* out      = (float*)d_out;                // 262144 out + loss1 + loss2
    float* loss_out = out + OUT_ELEMS;
    float* scale    = (float*)d_ws;                 // 512 floats
    float* scale_q  = scale + ROWS;                 // 512 floats

    vq_scale_kernel<<<ROWS / 8, 256, 0, stream>>>(feedback, scale, scale_q, loss_out);
    vq_main_kernel<<<ROWS, 256, 0, stream>>>(feedback, embed, scale, scale_q,
                                             out, loss_out);
}